// WRegLoss_35759897707024
// MI455X (gfx1250) — compile-verified
//
#include <hip/hip_runtime.h>

#define H_DIM   128
#define W_DIM   128
#define HW      16384
#define NT      256
#define EPT     (HW / NT)          // 64 elements per thread
#define CPT     (EPT / 4)          // 16 float4 chunks per thread
#define PXL_NUM 4096
#define KTH     12287u             // 0-indexed rank of threshold (HW - pxl_num - 1)
#define INV2S2  0.125f             // 1/(2*sigma^2), sigma=2

typedef int v4i __attribute__((ext_vector_type(4)));

// ---------- CDNA5 async global->LDS copy (ASYNCcnt path) ----------
__device__ __forceinline__ void async_cp16(const float* g, float* l) {
#if __has_builtin(__builtin_amdgcn_global_load_async_to_lds_b128)
  __builtin_amdgcn_global_load_async_to_lds_b128(
      (__attribute__((address_space(1))) v4i*)g,
      (__attribute__((address_space(3))) v4i*)l, 0, 0);
#else
  unsigned lo = (unsigned)(unsigned long long)
      (__attribute__((address_space(3))) void*)l;
  unsigned long long ga = (unsigned long long)g;
  asm volatile("global_load_async_to_lds_b128 %0, %1, off"
               :: "v"(lo), "v"(ga) : "memory");
#endif
}

__device__ __forceinline__ void wait_async0() {
#if __has_builtin(__builtin_amdgcn_s_wait_asynccnt)
  __builtin_amdgcn_s_wait_asynccnt(0);
#else
  asm volatile("s_wait_asynccnt 0" ::: "memory");
#endif
}

// Monotonic float<->key transform (handles any sign; target is nonneg anyway)
__device__ __forceinline__ unsigned float_key(float f) {
  unsigned u = __float_as_uint(f);
  return (u & 0x80000000u) ? ~u : (u | 0x80000000u);
}
__device__ __forceinline__ float key_float(unsigned k) {
  unsigned u = (k & 0x80000000u) ? (k & 0x7FFFFFFFu) : ~k;
  return __uint_as_float(u);
}

extern "C" __global__ void __launch_bounds__(NT)
wreg_row_kernel(const float* __restrict__ outp,
                const float* __restrict__ tgt,
                float* __restrict__ row_loss) {
  __shared__ __align__(16) float s_tgt[HW];   // 64 KB: one full target row
  __shared__ unsigned s_hist[256];
  __shared__ float    s_val[NT];
  __shared__ int      s_idx[NT];
  __shared__ unsigned s_sel[2];               // [0]=prefix key bits, [1]=remaining rank

  const int row = blockIdx.x;
  const int tid = threadIdx.x;
  const float* trow = tgt  + (size_t)row * HW;
  const float* prow = outp + (size_t)row * HW;

  // ---- Stage target row to LDS: 4096 x 16B async transfers ----
  #pragma unroll
  for (int j = 0; j < CPT; ++j) {
    int u = j * NT + tid;                     // 16-byte unit index
    async_cp16(trow + u * 4, s_tgt + u * 4);
  }
  if (tid == 0) { s_sel[0] = 0u; s_sel[1] = KTH; }
  wait_async0();
  __syncthreads();

  const float4* s_tgt4 = reinterpret_cast<const float4*>(s_tgt);

  // ---- Argmax (first-occurrence tie-break, matches jnp.argmax) ----
  float best = -__builtin_inff(); int bidx = 0;
  #pragma unroll 4
  for (int j = 0; j < CPT; ++j) {
    int c = tid + j * NT;
    float4 v = s_tgt4[c];                      // ds_load_b128
    int i = c * 4;                             // ascending order preserves 1st-occ.
    if (v.x > best) { best = v.x; bidx = i;     }
    if (v.y > best) { best = v.y; bidx = i + 1; }
    if (v.z > best) { best = v.z; bidx = i + 2; }
    if (v.w > best) { best = v.w; bidx = i + 3; }
  }
  s_val[tid] = best; s_idx[tid] = bidx;
  __syncthreads();
  for (int s = NT / 2; s > 0; s >>= 1) {
    if (tid < s) {
      float ov = s_val[tid + s]; int oi = s_idx[tid + s];
      float cv = s_val[tid];     int ci = s_idx[tid];
      if (ov > cv || (ov == cv && oi < ci)) { s_val[tid] = ov; s_idx[tid] = oi; }
    }
    __syncthreads();
  }
  const int   amax = s_idx[0];
  const float fcy  = (float)(amax >> 7);
  const float fcx  = (float)(amax & (W_DIM - 1));
  __syncthreads();

  // ---- Exact k-th smallest via 4-round byte radix select on LDS data ----
  unsigned prefix = 0u, pmask = 0u;
  for (int shift = 24; shift >= 0; shift -= 8) {
    s_hist[tid] = 0u;
    __syncthreads();
    for (int j = 0; j < CPT; ++j) {
      float4 v = s_tgt4[tid + j * NT];         // ds_load_b128
      unsigned k0 = float_key(v.x), k1 = float_key(v.y);
      unsigned k2 = float_key(v.z), k3 = float_key(v.w);
      if ((k0 & pmask) == prefix) atomicAdd(&s_hist[(k0 >> shift) & 0xFFu], 1u);
      if ((k1 & pmask) == prefix) atomicAdd(&s_hist[(k1 >> shift) & 0xFFu], 1u);
      if ((k2 & pmask) == prefix) atomicAdd(&s_hist[(k2 >> shift) & 0xFFu], 1u);
      if ((k3 & pmask) == prefix) atomicAdd(&s_hist[(k3 >> shift) & 0xFFu], 1u);
    }
    __syncthreads();
    if (tid == 0) {
      unsigned rem = s_sel[1];
      unsigned b = 0;
      for (; b < 255u; ++b) {
        unsigned c = s_hist[b];
        if (rem < c) break;
        rem -= c;
      }
      s_sel[0] = prefix | (b << shift);
      s_sel[1] = rem;
    }
    __syncthreads();
    prefix = s_sel[0];
    pmask |= (0xFFu << shift);
  }
  const float thresh = key_float(prefix);

  // ---- Masked MSE against re-rendered Gaussian (float4 global reads) ----
  const float4* prow4 = reinterpret_cast<const float4*>(prow);
  float acc = 0.0f;
  #pragma unroll 4
  for (int j = 0; j < CPT; ++j) {
    int c = tid + j * NT;
    float4 p = prow4[c];                       // global_load_b128, coalesced
    float4 t = s_tgt4[c];                      // ds_load_b128
    int i = c * 4;
    float dy  = (float)(i >> 7) - fcy;         // all 4 elems share the row
    float dy2 = dy * dy;
    float dx0 = (float)(i & (W_DIM - 1)) - fcx;
    float dx1 = dx0 + 1.0f, dx2 = dx0 + 2.0f, dx3 = dx0 + 3.0f;
    float g0 = __expf(-(dy2 + dx0 * dx0) * INV2S2);
    float g1 = __expf(-(dy2 + dx1 * dx1) * INV2S2);
    float g2 = __expf(-(dy2 + dx2 * dx2) * INV2S2);
    float g3 = __expf(-(dy2 + dx3 * dx3) * INV2S2);
    float d0 = p.x - g0, d1 = p.y - g1, d2 = p.z - g2, d3 = p.w - g3;
    acc += (t.x > thresh) ? d0 * d0 : 0.0f;
    acc += (t.y > thresh) ? d1 * d1 : 0.0f;
    acc += (t.z > thresh) ? d2 * d2 : 0.0f;
    acc += (t.w > thresh) ? d3 * d3 : 0.0f;
  }
  __syncthreads();
  s_val[tid] = acc;
  __syncthreads();
  for (int s = NT / 2; s > 0; s >>= 1) {
    if (tid < s) s_val[tid] += s_val[tid + s];
    __syncthreads();
  }
  if (tid == 0) row_loss[row] = s_val[0] * (1.0f / PXL_NUM);
}

extern "C" __global__ void __launch_bounds__(NT)
wreg_reduce_kernel(const float* __restrict__ row_loss, float* __restrict__ out, int n) {
  __shared__ float s[NT];
  float a = 0.0f;
  for (int i = threadIdx.x; i < n; i += NT) a += row_loss[i];
  s[threadIdx.x] = a;
  __syncthreads();
  for (int st = NT / 2; st > 0; st >>= 1) {
    if (threadIdx.x < st) s[threadIdx.x] += s[threadIdx.x + st];
    __syncthreads();
  }
  if (threadIdx.x == 0) out[0] = s[0] * (1.0f / (float)n);
}

extern "C" void kernel_launch(void* const* d_in, const int* in_sizes, int n_in,
                              void* d_out, int out_size, void* d_ws, size_t ws_size,
                              hipStream_t stream) {
  const float* output = (const float*)d_in[0];
  const float* target = (const float*)d_in[1];
  float* out = (float*)d_out;
  float* ws  = (float*)d_ws;

  const int rows = in_sizes[0] / HW;   // B*K = 1088

  wreg_row_kernel<<<dim3(rows), dim3(NT), 0, stream>>>(output, target, ws);
  wreg_reduce_kernel<<<dim3(1), dim3(NT), 0, stream>>>(ws, out, rows);
}